// AttentionPixelClassifier_76802605187428
// MI455X (gfx1250) — compile-verified
//
#include <hip/hip_runtime.h>
#include <math.h>

// ---- problem constants ----
#define C_   8
#define ES_  256
#define H_   256
#define W_   256
#define S_   (H_ * W_)          // 65536 pixels per channel
#define NH_  4
#define HD_  8
#define KSTEPS (ES_ / 4)        // 64 wmma k-steps of 4

typedef __attribute__((ext_vector_type(2))) float v2f;
typedef __attribute__((ext_vector_type(8))) float v8f;

// ---------------------------------------------------------------------------
// Prep kernel: one block per channel.
//   q(c)  = seed_emb(c, 0:255) @ Wq            (32-vector)
//   qw    = q * Wo[head] / sqrt(HD)
//   M(c)[e][2j]   = sum_h Wk[e,h8+2j]*qw[h8+2j]   + Wk[e,h8+2j+1]*qw[h8+2j+1]
//   M(c)[e][2j+1] = sum_h Wk[e,h8+2j+1]*qw[h8+2j] - Wk[e,h8+2j]  *qw[h8+2j+1]
// Written to ws pre-swizzled in WMMA A-fragment order (16x4 f32 A-matrix):
//   afrag[((c*64 + kstep)*32 + lane)*2 + v] = M[c][4*kstep + 2*(lane>>4) + v][lane&15]
// (cols 8..15 of M are zero padding so the 16x16 tile's upper cols vanish)
// ---------------------------------------------------------------------------
__global__ __launch_bounds__(256) void prep_M_kernel(
    const float* __restrict__ seed_emb,   // (8, 255)
    const float* __restrict__ Wq,         // (255, 32)
    const float* __restrict__ Wk,         // (256, 32)
    const float* __restrict__ Wo,         // (4, 1)
    float* __restrict__ afrag)            // ws: (8, 64, 32, 2) floats
{
  __shared__ float qw[NH_ * HD_];
  const int c = blockIdx.x;
  const int t = threadIdx.x;

  if (t < NH_ * HD_) {
    float s = 0.f;
    for (int i = 0; i < ES_ - 1; ++i)
      s = fmaf(seed_emb[c * (ES_ - 1) + i], Wq[i * (NH_ * HD_) + t], s);
    const int h = t >> 3;
    qw[t] = s * Wo[h] * 0.35355339059327373f;   // 1/sqrt(HD)
  }
  __syncthreads();

  const int e = t;                               // 0..255
  float m[8];
  #pragma unroll
  for (int j = 0; j < 4; ++j) {
    float a = 0.f, b = 0.f;
    #pragma unroll
    for (int h = 0; h < NH_; ++h) {
      const float w0 = Wk[e * 32 + h * 8 + 2 * j];
      const float w1 = Wk[e * 32 + h * 8 + 2 * j + 1];
      const float q0 = qw[h * 8 + 2 * j];
      const float q1 = qw[h * 8 + 2 * j + 1];
      a = fmaf(w0, q0, fmaf(w1, q1, a));
      b = fmaf(w1, q0, fmaf(-w0, q1, b));
    }
    m[2 * j]     = a;
    m[2 * j + 1] = b;
  }

  const int kstep = e >> 2;
  const int kk    = e & 3;
  const int v     = kk & 1;
  const int lhi   = kk >> 1;                     // lane-half holding this K
  #pragma unroll
  for (int n = 0; n < 16; ++n) {
    const int lane = lhi * 16 + n;
    afrag[(((size_t)c * KSTEPS + kstep) * 32 + lane) * 2 + v] = (n < 8) ? m[n] : 0.f;
  }
}

// ---------------------------------------------------------------------------
// Main kernel: one wave32 per 16-pixel tile.
//   D(16x16) = sum_k A(16x4, = M swizzled) x B(4x16, = 4 ES-planes of x)
//   After K=256: lanes 0..15 hold P[pixel][n=0..7] in acc[0..7].
//   Epilogue: logit = bo + sum_j cos(a_j)*P[2j] + sin(a_j)*P[2j+1]
// ---------------------------------------------------------------------------
__global__ __launch_bounds__(256) void attn_pix_kernel(
    const float* __restrict__ x,          // (8, 256, 256, 256)
    const float* __restrict__ afrag,      // ws swizzled M
    const float* __restrict__ bo,         // (1,)
    const float* __restrict__ freqs,      // (2,)
    float* __restrict__ out)              // (8*65536, 1)
{
  const int lane  = threadIdx.x & 31;
  const int wave  = threadIdx.x >> 5;
  const int w     = blockIdx.x * 8 + wave;       // global wave id, 0..32767
  const int c     = w >> 12;                     // 4096 tiles per channel
  const int tile  = w & 4095;
  const int pbase = tile << 4;                   // 16 consecutive pixels
  const int laneN = lane & 15;
  const int laneH = lane >> 4;                   // which K-half this lane holds

  // A fragments (M, L2-resident): one b64 per k-step, fully coalesced.
  const float* af = afrag + ((size_t)c * KSTEPS * 32 + lane) * 2;

  // B fragments (x, streamed once):
  //   VGPR0 = x[c][e0 + 2*laneH    ][pbase + laneN]
  //   VGPR1 = x[c][e0 + 2*laneH + 1][pbase + laneN]
  const float* bx = x + ((size_t)c << 24)                 // c * ES * H * W
                      + (size_t)(2 * laneH) * S_
                      + (size_t)pbase + laneN;

  v8f acc = {0.f, 0.f, 0.f, 0.f, 0.f, 0.f, 0.f, 0.f};

  #pragma unroll 8
  for (int ks = 0; ks < KSTEPS; ++ks) {
    v2f A = *(const v2f*)(af + (size_t)ks * 64);          // 32 lanes * 2 floats
    v2f B;
    B.x = __builtin_nontemporal_load(bx);                 // plane e0+2*laneH
    B.y = __builtin_nontemporal_load(bx + S_);            // plane e0+2*laneH+1
    bx += (size_t)4 * S_;                                 // next 4 ES planes
    // D(f32 16x16) = A(16x4 f32) * B(4x16 f32) + C
    acc = __builtin_amdgcn_wmma_f32_16x16x4_f32(
        /*neg_a=*/false, A, /*neg_b=*/false, B,
        /*c_mod=*/(short)0, acc, /*reuse_a=*/false, /*reuse_b=*/false);
  }

  // Epilogue: lanes 0..15 each own one pixel's 8 projections.
  if (laneH == 0) {
    const int s   = pbase + laneN;
    const int row = s >> 8;
    const int col = s & 255;
    const float py = (float)(row - H_ / 2);
    const float px = (float)(col - W_ / 2);
    const float f0 = freqs[0];
    const float f1 = freqs[1];
    const float a0 = py * f0, a1 = py * f1, a2 = px * f0, a3 = px * f1;
    float r = bo[0];
    r = fmaf(cosf(a0), acc[0], r);
    r = fmaf(sinf(a0), acc[1], r);
    r = fmaf(cosf(a1), acc[2], r);
    r = fmaf(sinf(a1), acc[3], r);
    r = fmaf(cosf(a2), acc[4], r);
    r = fmaf(sinf(a2), acc[5], r);
    r = fmaf(cosf(a3), acc[6], r);
    r = fmaf(sinf(a3), acc[7], r);
    out[((size_t)c << 16) + s] = r;
  }
}

extern "C" void kernel_launch(void* const* d_in, const int* in_sizes, int n_in,
                              void* d_out, int out_size, void* d_ws, size_t ws_size,
                              hipStream_t stream) {
  const float* x        = (const float*)d_in[0];
  const float* seed_emb = (const float*)d_in[1];
  const float* Wq       = (const float*)d_in[2];
  const float* Wk       = (const float*)d_in[3];
  const float* Wo       = (const float*)d_in[4];
  const float* bo       = (const float*)d_in[5];
  const float* freqs    = (const float*)d_in[6];
  float* out            = (float*)d_out;
  float* afrag          = (float*)d_ws;   // needs 8*64*32*2*4 = 128 KB

  prep_M_kernel<<<C_, 256, 0, stream>>>(seed_emb, Wq, Wk, Wo, afrag);

  // 8 channels * 4096 tiles = 32768 waves; 8 waves (256 threads) per block.
  attn_pix_kernel<<<4096, 256, 0, stream>>>(x, afrag, bo, freqs, out);
}